// VGAEEncoder_25598005084887
// MI455X (gfx1250) — compile-verified
//
#include <hip/hip_runtime.h>

typedef float v2f __attribute__((ext_vector_type(2)));
typedef float v8f __attribute__((ext_vector_type(8)));

static constexpr int IN_DIM = 256;
static constexpr int HID    = 128;

// ---------------------------------------------------------------------------
// native f32 atomic add (global_atomic_add_f32, no CAS loop)
// ---------------------------------------------------------------------------
__device__ __forceinline__ void atomAddF32(float* p, float v) {
    unsafeAtomicAdd(p, v);
}

// ---------------------------------------------------------------------------
// degree / normalization
// ---------------------------------------------------------------------------
__global__ __launch_bounds__(256) void k_init_cnt(unsigned* cnt, int n) {
    int i = blockIdx.x * blockDim.x + threadIdx.x;
    if (i < n) cnt[i] = 1u;                      // self-loop
}

__global__ __launch_bounds__(256) void k_count_deg(const int* __restrict__ dst,
                                                   unsigned* __restrict__ cnt, int e) {
    int i = blockIdx.x * blockDim.x + threadIdx.x;
    if (i < e) atomicAdd(&cnt[dst[i]], 1u);
}

__global__ __launch_bounds__(256) void k_dinv(const unsigned* __restrict__ cnt,
                                              float* __restrict__ dinv, int n) {
    int i = blockIdx.x * blockDim.x + threadIdx.x;
    if (i < n) dinv[i] = rsqrtf((float)cnt[i]);  // deg >= 1 always
}

// ---------------------------------------------------------------------------
// WMMA f32 GEMM:  D[N,128] = A[N,K] @ B[K,128]
//   - 32 rows (2 row-tiles) per block, staged via LDS (padded stride)
//   - 8 waves per block -> 8 column tiles of 16
//   - each wave keeps 2 accumulators; every B fragment feeds 2 WMMAs
//     (halves B VMEM ops per v_wmma vs the 16-row version)
//   - K/4 k-steps of v_wmma_f32_16x16x4_f32, exact f32 precision
//   - RELU:  fuse max(x,0) on the A tile (layer-2 reads pre-activation agg1)
//   - SPLITB: B is two [K,64] matrices (Wmu | Wls) side by side
// Fragment layouts per CDNA5 ISA 7.12.2:
//   A 16x4 f32 : lanes 0-15 rows M, v[0]=K+0,v[1]=K+1; lanes 16-31: K+2,K+3
//   B 4x16 f32 : lanes 0-15 cols N, v[0]=K+0,v[1]=K+1; lanes 16-31: K+2,K+3
//   C/D 16x16  : lane 0-15 -> N=lane, M=vgpr(0..7); lane 16-31 -> M=8+vgpr
// ---------------------------------------------------------------------------
template <int K, bool RELU, bool SPLITB>
__global__ __launch_bounds__(256, 2)
void k_gemm_wmma(const float* __restrict__ A,
                 const float* __restrict__ B0,
                 const float* __restrict__ B1,
                 float* __restrict__ D)
{
    constexpr int MB = 32;                   // rows per block (N % 32 == 0)
    constexpr int LK = K + 4;                // padded LDS row stride (banks)
    __shared__ float As[MB * LK];

    const int tid  = threadIdx.x;
    const int wave = tid >> 5;
    const int lane = tid & 31;
    const int row0 = blockIdx.x * MB;

    // cooperative load of the 32xK A tile (contiguous 32*K floats, b128 loads)
    const float* Atile = A + (size_t)row0 * K;
#pragma unroll
    for (int c = 0; c < MB * K / (4 * 256); ++c) {
        const int f = (tid + c * 256) * 4;   // flat float index in tile
        float4 v = *(const float4*)(Atile + f);
        if (RELU) {
            v.x = fmaxf(v.x, 0.0f); v.y = fmaxf(v.y, 0.0f);
            v.z = fmaxf(v.z, 0.0f); v.w = fmaxf(v.w, 0.0f);
        }
        const int r = f / K, k = f % K;      // k is a multiple of 4
        *(float4*)(&As[r * LK + k]) = v;
    }
    __syncthreads();

    const int col0 = wave << 4;
    const int mrow = lane & 15;
    const int kofs = (lane < 16) ? 0 : 2;

    const float* Bp;
    int bcol, ldb;
    if (SPLITB) {
        ldb = 64;
        if (col0 < 64) { Bp = B0; bcol = col0 + mrow; }
        else           { Bp = B1; bcol = col0 - 64 + mrow; }
    } else {
        ldb = HID; Bp = B0; bcol = col0 + mrow;
    }

    v8f acc0 = {};
    v8f acc1 = {};
#pragma unroll 4
    for (int k0 = 0; k0 < K; k0 += 4) {
        const int kk = k0 + kofs;
        // one B fragment, reused by both row-tiles
        v2f b;
        b.x = Bp[(size_t)kk * ldb + bcol];
        b.y = Bp[(size_t)(kk + 1) * ldb + bcol];
        // A fragments for row-tile 0 and row-tile 1 (ds_load_b64 each)
        const float2 av0 = *(const float2*)(&As[mrow * LK + kk]);
        const float2 av1 = *(const float2*)(&As[(16 + mrow) * LK + kk]);
        v2f a0, a1;
        a0.x = av0.x; a0.y = av0.y;
        a1.x = av1.x; a1.y = av1.y;
        acc0 = __builtin_amdgcn_wmma_f32_16x16x4_f32(false, a0, false, b,
                                                     (short)0, acc0, false, false);
        acc1 = __builtin_amdgcn_wmma_f32_16x16x4_f32(false, a1, false, b,
                                                     (short)0, acc1, false, false);
    }

    const int n  = col0 + mrow;
    const int m0 = row0 + ((lane < 16) ? 0 : 8);
#pragma unroll
    for (int r = 0; r < 8; ++r)
        D[(size_t)(m0 + r) * HID + n] = acc0[r];
#pragma unroll
    for (int r = 0; r < 8; ++r)
        D[(size_t)(m0 + 16 + r) * HID + n] = acc1[r];
}

// ---------------------------------------------------------------------------
// agg1[i,c] = b1[c] + dinv[i]^2 * t1[i,c]      (bias + self-loop fused)
// one wave per node, lane = 4 channels
// ---------------------------------------------------------------------------
__global__ __launch_bounds__(256)
void k_init_agg1(const float* __restrict__ t1, const float* __restrict__ dinv,
                 const float* __restrict__ b1, float* __restrict__ agg, int n)
{
    int tid = blockIdx.x * blockDim.x + threadIdx.x;
    int i = tid >> 5;
    if (i >= n) return;
    int c0 = (tid & 31) << 2;
    float di = dinv[i], sl = di * di;
    float4 v  = *(const float4*)(t1 + (size_t)i * HID + c0);
    float4 bb = *(const float4*)(b1 + c0);
    float4 o  = { bb.x + sl * v.x, bb.y + sl * v.y,
                  bb.z + sl * v.z, bb.w + sl * v.w };
    *(float4*)(agg + (size_t)i * HID + c0) = o;
}

// ---------------------------------------------------------------------------
// edge scatter: agg[dst,:] += dinv[src]*dinv[dst] * t[src,:]
// one wave per edge, lane = 4 channels (coalesced b128 gather + f32 atomics)
// ---------------------------------------------------------------------------
__global__ __launch_bounds__(256)
void k_scatter(const int* __restrict__ src, const int* __restrict__ dst,
               const float* __restrict__ dinv, const float* __restrict__ t,
               float* __restrict__ agg, int e)
{
    int tid  = blockIdx.x * blockDim.x + threadIdx.x;
    int eidx = tid >> 5;
    if (eidx >= e) return;
    int lane = tid & 31;
    int s = src[eidx], d = dst[eidx];
    float nrm = dinv[s] * dinv[d];
    int c0 = lane << 2;
    const float4 v = *(const float4*)(t + (size_t)s * HID + c0);
    float* p = agg + (size_t)d * HID + c0;
    atomAddF32(p + 0, nrm * v.x);
    atomAddF32(p + 1, nrm * v.y);
    atomAddF32(p + 2, nrm * v.z);
    atomAddF32(p + 3, nrm * v.w);
}

// ---------------------------------------------------------------------------
// layer-2 init straight into d_out with the (mu | logstd) split layout:
//   c<64  -> mu      at out[i*64 + c]
//   c>=64 -> logstd  at out[N*64 + i*64 + (c-64)]
// ---------------------------------------------------------------------------
__global__ __launch_bounds__(256)
void k_init_out(const float* __restrict__ t2, const float* __restrict__ dinv,
                const float* __restrict__ bmu, const float* __restrict__ bls,
                float* __restrict__ out, int n)
{
    int tid = blockIdx.x * blockDim.x + threadIdx.x;
    int i = tid >> 5;
    if (i >= n) return;
    int c0 = (tid & 31) << 2;
    float di = dinv[i], sl = di * di;
    float4 v = *(const float4*)(t2 + (size_t)i * HID + c0);
    const float* bsrc;
    float* dp;
    if (c0 < 64) { bsrc = bmu + c0;        dp = out + (size_t)i * 64 + c0; }
    else         { bsrc = bls + (c0 - 64); dp = out + (size_t)n * 64 + (size_t)i * 64 + (c0 - 64); }
    float4 bb = *(const float4*)bsrc;
    float4 o  = { bb.x + sl * v.x, bb.y + sl * v.y,
                  bb.z + sl * v.z, bb.w + sl * v.w };
    *(float4*)dp = o;
}

__global__ __launch_bounds__(256)
void k_scatter_out(const int* __restrict__ src, const int* __restrict__ dst,
                   const float* __restrict__ dinv, const float* __restrict__ t2,
                   float* __restrict__ out, int n, int e)
{
    int tid  = blockIdx.x * blockDim.x + threadIdx.x;
    int eidx = tid >> 5;
    if (eidx >= e) return;
    int lane = tid & 31;
    int s = src[eidx], d = dst[eidx];
    float nrm = dinv[s] * dinv[d];
    int c0 = lane << 2;
    const float4 v = *(const float4*)(t2 + (size_t)s * HID + c0);
    float* p = (c0 < 64) ? (out + (size_t)d * 64 + c0)
                         : (out + (size_t)n * 64 + (size_t)d * 64 + (c0 - 64));
    atomAddF32(p + 0, nrm * v.x);
    atomAddF32(p + 1, nrm * v.y);
    atomAddF32(p + 2, nrm * v.z);
    atomAddF32(p + 3, nrm * v.w);
}

// ---------------------------------------------------------------------------
// launch
// ---------------------------------------------------------------------------
extern "C" void kernel_launch(void* const* d_in, const int* in_sizes, int n_in,
                              void* d_out, int out_size, void* d_ws, size_t ws_size,
                              hipStream_t stream)
{
    const float* x   = (const float*)d_in[0];
    const int*   ei  = (const int*)d_in[1];   // [2,E]: row0 = src, row1 = dst
    const float* W1  = (const float*)d_in[2];
    const float* b1  = (const float*)d_in[3];
    const float* Wmu = (const float*)d_in[4];
    const float* bmu = (const float*)d_in[5];
    const float* Wls = (const float*)d_in[6];
    const float* bls = (const float*)d_in[7];

    const int N = in_sizes[0] / IN_DIM;       // 100000 (divisible by 32)
    const int E = in_sizes[1] / 2;            // 1600000

    float*    ws   = (float*)d_ws;
    float*    dinv = ws;                                  // N f32
    unsigned* cnt  = (unsigned*)(ws + N);                 // N u32
    float*    bufA = ws + 2 * (size_t)N;                  // t1, reused as t2
    float*    bufB = bufA + (size_t)N * HID;              // agg1 (pre-ReLU h)
    float*    out  = (float*)d_out;

    const int* srcp = ei;
    const int* dstp = ei + E;

    const int nodeBlk = (N + 255) / 256;
    const int edgeBlk = (E + 255) / 256;
    const int nodeWaveBlk = (int)(((size_t)N * 32 + 255) / 256);
    const int edgeWaveBlk = (int)(((size_t)E * 32 + 255) / 256);

    // 1. symmetric degree normalization
    k_init_cnt <<<nodeBlk, 256, 0, stream>>>(cnt, N);
    k_count_deg<<<edgeBlk, 256, 0, stream>>>(dstp, cnt, E);
    k_dinv     <<<nodeBlk, 256, 0, stream>>>(cnt, dinv, N);

    // 2. t1 = x @ W1                        (WMMA f32, exact)
    k_gemm_wmma<IN_DIM, false, false><<<N / 32, 256, 0, stream>>>(x, W1, nullptr, bufA);

    // 3. agg1 = b1 + dinv^2*t1 + scatter-add over edges
    k_init_agg1<<<nodeWaveBlk, 256, 0, stream>>>(bufA, dinv, b1, bufB, N);
    k_scatter  <<<edgeWaveBlk, 256, 0, stream>>>(srcp, dstp, dinv, bufA, bufB, E);

    // 4. t2 = relu(agg1) @ [Wmu | Wls]      (ReLU fused into A-tile load)
    k_gemm_wmma<HID, true, true><<<N / 32, 256, 0, stream>>>(bufB, Wmu, Wls, bufA);

    // 5. out = bias + dinv^2*t2 + scatter-add, written in (mu, logstd) layout
    k_init_out   <<<nodeWaveBlk, 256, 0, stream>>>(bufA, dinv, bmu, bls, out, N);
    k_scatter_out<<<edgeWaveBlk, 256, 0, stream>>>(srcp, dstp, dinv, bufA, out, N, E);
}